// GATNodePairScorer_74165495267774
// MI455X (gfx1250) — compile-verified
//
#include <hip/hip_runtime.h>
#include <hip/hip_bf16.h>

// ---------------------------------------------------------------------------
// GAT node-pair scorer for MI455X (gfx1250, wave32).
// Dense GEMMs use V_WMMA_F32_16X16X4_F32 (fp32 keeps reference numerics; the
// GEMMs are ~5 GFLOP total and negligible next to the edge scatter phase,
// whose working set (g: 51MB, acc: 51MB) fits entirely in the 192MB L2).
// ---------------------------------------------------------------------------

#define N_NODES 50000
#define N_EDGES 800000
#define N_PAIRS 100000
#define IN_DIM  32
#define HID     64
#define HEADS   4
#define OUTC    64

typedef __attribute__((ext_vector_type(2))) float v2f;
typedef __attribute__((ext_vector_type(8))) float v8f;

__device__ __forceinline__ v8f wmma_f32x4(v2f a, v2f b, v8f c) {
  // D = A(16x4) * B(4x16) + C(16x16), fp32, one wave.
  return __builtin_amdgcn_wmma_f32_16x16x4_f32(false, a, false, b,
                                               (short)0, c, false, false);
}

__device__ __forceinline__ float leaky(float v, float slope) {
  return v >= 0.0f ? v : slope * v;
}

// order-preserving float<->uint mapping for atomic max
__device__ __forceinline__ unsigned flt_to_ord(float f) {
  unsigned u = __float_as_uint(f);
  return (u & 0x80000000u) ? ~u : (u | 0x80000000u);
}
__device__ __forceinline__ float ord_to_flt(unsigned o) {
  unsigned u = (o & 0x80000000u) ? (o & 0x7fffffffu) : ~o;
  return __uint_as_float(u);
}
#define ORD_NEG_INF 0x007FFFFFu  // flt_to_ord(-inf)

// ---------------------------------------------------------------------------
// Fragment loaders per CDNA5 ISA VGPR layouts (CDNA5 ISA §7.12.2).
// A 16x4 (MxK): lanes 0-15 hold K={k,k+1}, lanes 16-31 hold K={k+2,k+3},
//               M = lane&15, 2 VGPRs per lane.
// B 4x16 (KxN): row striped across lanes; lanes 0-15 K={k,k+1},
//               lanes 16-31 K={k+2,k+3}, N = n0 + (lane&15).
// C/D 16x16:    VGPR j -> row j (lanes 0-15) / row 8+j (lanes 16-31),
//               col = n0 + (lane&15).
// ---------------------------------------------------------------------------
__device__ __forceinline__ v2f load_a_frag(const float* __restrict__ A,
                                           int M, int K, int m0, int k, int lane) {
  int m = m0 + (lane & 15);
  if (m >= M) m = M - 1;                 // clamp; stores are masked
  int klo = k + ((lane >> 4) << 1);
  const float* p = A + (size_t)m * K + klo;
  v2f a; a.x = p[0]; a.y = p[1];
  return a;
}

__device__ __forceinline__ v2f load_b_frag(const float* __restrict__ B,
                                           int N, int k, int n0, int lane) {
  int n  = n0 + (lane & 15);
  int kb = k + ((lane >> 4) << 1);
  v2f b; b.x = B[(size_t)kb * N + n]; b.y = B[(size_t)(kb + 1) * N + n];
  return b;
}

// ---------------------------------------------------------------------------
// Generic 16-row x 64-col WMMA GEMM tile per wave:  C = act(A@B + bias)
// EPI: 0 = none, 1 = bias + leaky(0.01)
// ---------------------------------------------------------------------------
template<int K, int EPI>
__global__ __launch_bounds__(32)
void gemm16_kernel(const float* __restrict__ A, const float* __restrict__ B,
                   const float* __restrict__ bias, float* __restrict__ C,
                   int M, int N) {
  const int lane = threadIdx.x & 31;
  const int m0 = blockIdx.x * 16;
  const int n0 = blockIdx.y * 64;
  v8f acc[4] = {};
#pragma unroll
  for (int k = 0; k < K; k += 4) {
    v2f a = load_a_frag(A, M, K, m0, k, lane);
#pragma unroll
    for (int t = 0; t < 4; ++t) {
      v2f b = load_b_frag(B, N, k, n0 + t * 16, lane);
      acc[t] = wmma_f32x4(a, b, acc[t]);
    }
  }
  const int mb = m0 + ((lane >> 4) << 3);
#pragma unroll
  for (int t = 0; t < 4; ++t) {
    const int n = n0 + t * 16 + (lane & 15);
    const float bv = (EPI == 1) ? bias[n] : 0.0f;
#pragma unroll
    for (int j = 0; j < 8; ++j) {
      int m = mb + j;
      if (m < M) {
        float v = acc[t][j];
        if (EPI == 1) v = leaky(v + bv, 0.01f);
        C[(size_t)m * N + n] = v;
      }
    }
  }
}

// ---------------------------------------------------------------------------
// Per-(node,head) attention logits; att[4][128] staged in LDS.
// a_dst = <g[n,h,:], att[h,0:64]>, a_src = <g[n,h,:], att[h,64:128]>
// ---------------------------------------------------------------------------
__global__ void node_att_kernel(const float* __restrict__ g,
                                const float* __restrict__ att,
                                float* __restrict__ a_dst,
                                float* __restrict__ a_src, int Nn) {
  __shared__ float s_att[HEADS * 2 * OUTC];
  for (int i = threadIdx.x; i < HEADS * 2 * OUTC; i += blockDim.x)
    s_att[i] = att[i];
  __syncthreads();
  int t = blockIdx.x * blockDim.x + threadIdx.x;  // node*4 + head
  if (t >= Nn * HEADS) return;
  const int n = t >> 2, hh = t & 3;
  const float4* gp = (const float4*)(g + (size_t)n * (HEADS * OUTC) + hh * OUTC);
  float sd = 0.0f, ss = 0.0f;
#pragma unroll
  for (int i = 0; i < OUTC / 4; ++i) {
    float4 v = gp[i];
    const float* ad = &s_att[hh * 128 + i * 4];
    const float* as = &s_att[hh * 128 + 64 + i * 4];
    sd += v.x * ad[0] + v.y * ad[1] + v.z * ad[2] + v.w * ad[3];
    ss += v.x * as[0] + v.y * as[1] + v.z * as[2] + v.w * as[3];
  }
  a_dst[t] = sd;
  a_src[t] = ss;
}

// ---------------------------------------------------------------------------
// Per-layer scratch init: m_ord=-inf, denom=0, acc=0
// ---------------------------------------------------------------------------
__global__ void init_layer_kernel(unsigned* __restrict__ m_ord,
                                  float* __restrict__ denom,
                                  float* __restrict__ acc, int Nn) {
  int t = blockIdx.x * blockDim.x + threadIdx.x;
  if (t < Nn * HEADS) { m_ord[t] = ORD_NEG_INF; denom[t] = 0.0f; }
  if (t < Nn * HEADS * OUTC) acc[t] = 0.0f;
}

__device__ __forceinline__ void edge_sd(const int* __restrict__ ei, int e,
                                        int nE, int& s, int& d) {
  if (e < nE) { s = ei[e]; d = ei[nE + e]; }
  else        { s = d = e - nE; }         // self-loop
}

// Pass 1: e = leaky(a_dst[d]+a_src[s], 0.2); segment-max via ordered-uint atomic
__global__ void edge_pass1_kernel(const int* __restrict__ ei,
                                  const float* __restrict__ a_dst,
                                  const float* __restrict__ a_src,
                                  float* __restrict__ e_buf,
                                  unsigned* __restrict__ m_ord,
                                  int nE, int nTot) {
  int t = blockIdx.x * blockDim.x + threadIdx.x;  // edge*4 + head
  if (t >= nTot * HEADS) return;
  int e = t >> 2, hh = t & 3, s, d;
  edge_sd(ei, e, nE, s, d);
  float v = leaky(a_dst[d * HEADS + hh] + a_src[s * HEADS + hh], 0.2f);
  e_buf[t] = v;
  atomicMax(&m_ord[d * HEADS + hh], flt_to_ord(v));
}

// Pass 2: ex = exp(e - m[d]); segment-sum denom
__global__ void edge_pass2_kernel(const int* __restrict__ ei,
                                  float* __restrict__ e_buf,
                                  const unsigned* __restrict__ m_ord,
                                  float* __restrict__ denom,
                                  int nE, int nTot) {
  int t = blockIdx.x * blockDim.x + threadIdx.x;
  if (t >= nTot * HEADS) return;
  int e = t >> 2, hh = t & 3, s, d;
  edge_sd(ei, e, nE, s, d);
  float m = ord_to_flt(m_ord[d * HEADS + hh]);
  float ex = expf(e_buf[t] - m);
  e_buf[t] = ex;
  atomicAdd(&denom[d * HEADS + hh], ex);
}

// Pass 3: acc[d,h,:] += (ex/denom[d,h]) * g[s,h,:]  (float4 gather, 4 atomics)
__global__ void edge_pass3_kernel(const int* __restrict__ ei,
                                  const float* __restrict__ g,
                                  const float* __restrict__ e_buf,
                                  const float* __restrict__ denom,
                                  float* __restrict__ acc,
                                  int nE, int nTot) {
  int t = blockIdx.x * blockDim.x + threadIdx.x;  // edge*64 + (head*16 + c4)
  if (t >= nTot * 64) return;
  int e = t >> 6, r = t & 63;
  int hh = r >> 4, c4 = r & 15;
  int s, d;
  edge_sd(ei, e, nE, s, d);
  float alpha = e_buf[e * HEADS + hh] / denom[d * HEADS + hh];
  const float4 gv = *(const float4*)(g + (size_t)s * 256 + hh * OUTC + c4 * 4);
  float* ap = acc + (size_t)d * 256 + hh * OUTC + c4 * 4;
  atomicAdd(ap + 0, alpha * gv.x);
  atomicAdd(ap + 1, alpha * gv.y);
  atomicAdd(ap + 2, alpha * gv.z);
  atomicAdd(ap + 3, alpha * gv.w);
}

// head-mean + bias (+ optional leaky 0.01)
template<int LEAKY>
__global__ void finalize_kernel(const float* __restrict__ acc,
                                const float* __restrict__ bias,
                                float* __restrict__ hout, int Nn) {
  int t = blockIdx.x * blockDim.x + threadIdx.x;  // node*64 + o
  if (t >= Nn * OUTC) return;
  int n = t >> 6, o = t & 63;
  const float* ap = acc + (size_t)n * 256 + o;
  float v = 0.25f * (ap[0] + ap[64] + ap[128] + ap[192]) + bias[o];
  if (LEAKY) v = leaky(v, 0.01f);
  hout[t] = v;
}

// ---------------------------------------------------------------------------
// Fused pair scorer: one wave = 16 pairs.
// s = leaky([h[i]|h[j]] @ Ws1 + bs1, 0.01);  score = s @ Ws2 + bs2
// A-frag gathers directly from h rows (K=128: k<64 -> h[i], else h[j]).
// Ws2-dot folded into epilogue; cross-lane shfl_xor reduction per row.
// ---------------------------------------------------------------------------
__global__ __launch_bounds__(32)
void pair_score_kernel(const float* __restrict__ h,
                       const int* __restrict__ ni, const int* __restrict__ nj,
                       const float* __restrict__ Ws1, const float* __restrict__ bs1,
                       const float* __restrict__ Ws2, const float* __restrict__ bs2,
                       float* __restrict__ out, int P) {
  const int lane = threadIdx.x & 31;
  const int m0 = blockIdx.x * 16;
  int row = m0 + (lane & 15);
  if (row >= P) row = P - 1;
  const int node_i = ni[row], node_j = nj[row];

  v8f acc[4] = {};
#pragma unroll
  for (int k = 0; k < 2 * HID; k += 4) {
    int klo = k + ((lane >> 4) << 1);
    int node = (klo < HID) ? node_i : node_j;
    int col  = klo & (HID - 1);
    const float* p = h + (size_t)node * HID + col;
    v2f a; a.x = p[0]; a.y = p[1];
#pragma unroll
    for (int t = 0; t < 4; ++t) {
      int n = t * 16 + (lane & 15);
      v2f b; b.x = Ws1[(size_t)klo * HID + n];
             b.y = Ws1[(size_t)(klo + 1) * HID + n];
      acc[t] = wmma_f32x4(a, b, acc[t]);
    }
  }

  // epilogue: bias + leaky + dot with Ws2, per-lane partial over its columns
  float part[8] = {0, 0, 0, 0, 0, 0, 0, 0};
#pragma unroll
  for (int t = 0; t < 4; ++t) {
    const int n = t * 16 + (lane & 15);
    const float b1 = bs1[n];
    const float w2 = Ws2[n];
#pragma unroll
    for (int j = 0; j < 8; ++j)
      part[j] += leaky(acc[t][j] + b1, 0.01f) * w2;
  }
  // reduce over 16 lanes within each half-wave (rows 0-7 / 8-15 of the tile)
#pragma unroll
  for (int j = 0; j < 8; ++j)
#pragma unroll
    for (int m = 8; m >= 1; m >>= 1)
      part[j] += __shfl_xor(part[j], m, 32);

  if ((lane & 15) == 0) {
    const int mb = m0 + ((lane >> 4) << 3);
    const float b2 = bs2[0];
#pragma unroll
    for (int j = 0; j < 8; ++j) {
      int r = mb + j;
      if (r < P) out[r] = part[j] + b2;
    }
  }
}

// ---------------------------------------------------------------------------
// Host-side orchestration
// ---------------------------------------------------------------------------
static void run_gat_layer(const float* hin, const float* Wg, const float* att,
                          const float* bg, float* g, float* a_dst, float* a_src,
                          unsigned* m_ord, float* denom, float* e_buf, float* acc,
                          float* hout, const int* ei, int Nn, int nE, int nTot,
                          int leaky_out, hipStream_t stream) {
  dim3 gtile((Nn + 15) / 16, 4);  // N=256 -> 4 column tiles of 64
  gemm16_kernel<HID, 0><<<gtile, 32, 0, stream>>>(hin, Wg, nullptr, g, Nn, 256);

  int ta = Nn * HEADS;
  node_att_kernel<<<(ta + 255) / 256, 256, 0, stream>>>(g, att, a_dst, a_src, Nn);

  int ti = Nn * HEADS * OUTC;
  init_layer_kernel<<<(ti + 255) / 256, 256, 0, stream>>>(m_ord, denom, acc, Nn);

  int te = nTot * HEADS;
  edge_pass1_kernel<<<(te + 255) / 256, 256, 0, stream>>>(ei, a_dst, a_src, e_buf,
                                                          m_ord, nE, nTot);
  edge_pass2_kernel<<<(te + 255) / 256, 256, 0, stream>>>(ei, e_buf, m_ord, denom,
                                                          nE, nTot);
  long long tm = (long long)nTot * 64;
  edge_pass3_kernel<<<(unsigned)((tm + 255) / 256), 256, 0, stream>>>(
      ei, g, e_buf, denom, acc, nE, nTot);

  int tf = Nn * OUTC;
  if (leaky_out)
    finalize_kernel<1><<<(tf + 255) / 256, 256, 0, stream>>>(acc, bg, hout, Nn);
  else
    finalize_kernel<0><<<(tf + 255) / 256, 256, 0, stream>>>(acc, bg, hout, Nn);
}

extern "C" void kernel_launch(void* const* d_in, const int* in_sizes, int n_in,
                              void* d_out, int out_size, void* d_ws, size_t ws_size,
                              hipStream_t stream) {
  const float* x    = (const float*)d_in[0];
  const int*   ei   = (const int*)  d_in[1];
  const int*   ni   = (const int*)  d_in[2];
  const int*   nj   = (const int*)  d_in[3];
  const float* W_in = (const float*)d_in[4];
  const float* b_in = (const float*)d_in[5];
  const float* Wg0  = (const float*)d_in[6];
  const float* att0 = (const float*)d_in[7];
  const float* bg0  = (const float*)d_in[8];
  const float* Wg1  = (const float*)d_in[9];
  const float* att1 = (const float*)d_in[10];
  const float* bg1  = (const float*)d_in[11];
  const float* Ws1  = (const float*)d_in[12];
  const float* bs1  = (const float*)d_in[13];
  const float* Ws2  = (const float*)d_in[14];
  const float* bs2  = (const float*)d_in[15];
  float* out = (float*)d_out;

  const int Nn   = in_sizes[0] / IN_DIM;   // 50000
  const int nE   = in_sizes[1] / 2;        // 800000
  const int P    = in_sizes[2];            // 100000
  const int nTot = nE + Nn;                // + self-loops

  // workspace layout (floats)
  float* ws = (float*)d_ws;
  float*    h0    = ws;                    ws += (size_t)Nn * HID;
  float*    h1    = ws;                    ws += (size_t)Nn * HID;
  float*    g     = ws;                    ws += (size_t)Nn * HEADS * OUTC;
  float*    adst  = ws;                    ws += (size_t)Nn * HEADS;
  float*    asrc  = ws;                    ws += (size_t)Nn * HEADS;
  unsigned* mord  = (unsigned*)ws;         ws += (size_t)Nn * HEADS;
  float*    den   = ws;                    ws += (size_t)Nn * HEADS;
  float*    ebuf  = ws;                    ws += (size_t)nTot * HEADS;
  float*    acc   = ws;                    ws += (size_t)Nn * HEADS * OUTC;

  // h0 = leaky(x @ W_in + b_in, 0.01)   [M=Nn, K=32, N=64]
  dim3 g_in((Nn + 15) / 16, 1);
  gemm16_kernel<IN_DIM, 1><<<g_in, 32, 0, stream>>>(x, W_in, b_in, h0, Nn, HID);

  // GAT layer 0 (leaky output), layer 1 (plain output)
  run_gat_layer(h0, Wg0, att0, bg0, g, adst, asrc, mord, den, ebuf, acc, h1,
                ei, Nn, nE, nTot, /*leaky_out=*/1, stream);
  run_gat_layer(h1, Wg1, att1, bg1, g, adst, asrc, mord, den, ebuf, acc, h0,
                ei, Nn, nE, nTot, /*leaky_out=*/0, stream);

  // pair scorer
  pair_score_kernel<<<(P + 15) / 16, 32, 0, stream>>>(h0, ni, nj, Ws1, bs1,
                                                      Ws2, bs2, out, P);
}